// Attention_19516331393201
// MI455X (gfx1250) — compile-verified
//
#include <hip/hip_runtime.h>
#include <hip/hip_bf16.h>

typedef float v2f __attribute__((ext_vector_type(2)));
typedef float v8f __attribute__((ext_vector_type(8)));

#define RADIUS   50
#define NEMB     (2 * RADIUS + 1)   // 101
#define HEADS    4
#define CHANNELS 256
#define DIM      64                 // CHANNELS / HEADS
#define TLEN     1024
#define BATCH    8
#define BN_EPS   1e-5f

// ---------------------------------------------------------------------------
// GEMM: Y[b][o][t] = sum_c W[o][c] * X[b][c][t] + bias[o]
// M = 256 (o), N = 1024 (t), K = 256 (c), per batch b.
// fp32 WMMA: V_WMMA_F32_16X16X4_F32.  Wave computes a 32x32 output tile
// (2x2 of 16x16), block = 8 waves -> 128(M) x 64(N).  Grid (N/64, M/128, B).
// A 16x4 fragment (ISA 7.12.2): lanes 0-15 hold M=lane, K=k+0..1;
// lanes 16-31 hold M=lane-16, K=k+2..3.  B mirrors with N across lanes.
// C/D: VGPR r = row r (lanes 0-15) / row r+8 (lanes 16-31), lane&15 = col.
// ---------------------------------------------------------------------------
__global__ __launch_bounds__(256)
void proj_gemm_wmma(const float* __restrict__ W, const float* __restrict__ bias,
                    const float* __restrict__ X, float* __restrict__ Y)
{
    const int lane = threadIdx.x & 31;
    const int wave = threadIdx.x >> 5;
    const int hl   = lane >> 4;       // half-wave select (K split)
    const int l16  = lane & 15;
    const int b    = blockIdx.z;

    const int Mbase = blockIdx.y * 128 + (wave >> 1) * 32;
    const int Nbase = blockIdx.x * 64  + (wave & 1) * 32;

    const float* Xb = X + (size_t)b * CHANNELS * TLEN;
    float*       Yb = Y + (size_t)b * CHANNELS * TLEN;

    v8f acc[2][2];
    const v8f vzero = {0.f, 0.f, 0.f, 0.f, 0.f, 0.f, 0.f, 0.f};
#pragma unroll
    for (int i = 0; i < 2; ++i)
#pragma unroll
        for (int j = 0; j < 2; ++j) acc[i][j] = vzero;

    for (int k = 0; k < CHANNELS; k += 4) {
        const int kk = k + 2 * hl;

        v2f afrag[2];
#pragma unroll
        for (int i = 0; i < 2; ++i) {
            const int m = Mbase + i * 16 + l16;
            // W[m][kk], W[m][kk+1] contiguous -> 8-byte load
            afrag[i] = *(const v2f*)(W + (size_t)m * CHANNELS + kk);
        }

        v2f bfrag[2];
#pragma unroll
        for (int j = 0; j < 2; ++j) {
            const int t = Nbase + j * 16 + l16;
            v2f bb;
            bb.x = Xb[(size_t)kk * TLEN + t];
            bb.y = Xb[(size_t)(kk + 1) * TLEN + t];
            bfrag[j] = bb;
        }

#pragma unroll
        for (int i = 0; i < 2; ++i)
#pragma unroll
            for (int j = 0; j < 2; ++j)
                acc[i][j] = __builtin_amdgcn_wmma_f32_16x16x4_f32(
                    /*neg_a=*/false, afrag[i], /*neg_b=*/false, bfrag[j],
                    /*c_mod=*/(short)0, acc[i][j],
                    /*reuse_a=*/false, /*reuse_b=*/false);
    }

#pragma unroll
    for (int i = 0; i < 2; ++i) {
#pragma unroll
        for (int j = 0; j < 2; ++j) {
            const int t = Nbase + j * 16 + l16;
#pragma unroll
            for (int r = 0; r < 8; ++r) {
                const int m = Mbase + i * 16 + r + 8 * hl;
                Yb[(size_t)m * TLEN + t] = acc[i][j][r] + bias[m];
            }
        }
    }
}

// ---------------------------------------------------------------------------
// Banded relative-position attention.  One wave per (b,h,t).
// Block = 128 threads = 4 waves; emb staged once per block in LDS.
// ---------------------------------------------------------------------------
__global__ __launch_bounds__(128)
void band_attention(const float* __restrict__ Q, const float* __restrict__ Kp,
                    const float* __restrict__ Cn, const float* __restrict__ emb,
                    float* __restrict__ O)
{
    __shared__ float s_emb[NEMB * DIM];   // 25856 B
    __shared__ float s_q[4][DIM];
    __shared__ float s_w[4][128];

    for (int i = threadIdx.x; i < NEMB * DIM; i += blockDim.x) s_emb[i] = emb[i];
    __syncthreads();

    const int lane = threadIdx.x & 31;
    const int wave = threadIdx.x >> 5;
    const int idx  = blockIdx.x * 4 + wave;   // linear (b,h,t)
    const int t = idx & (TLEN - 1);
    const int h = (idx >> 10) & (HEADS - 1);
    const int b = idx >> 12;

    const size_t base = (size_t)b * CHANNELS * TLEN + (size_t)h * DIM * TLEN;
    const float* Qb = Q + base;    // (c, t), c in [0, DIM)
    const float* Kb = Kp + base;
    const float* Cb = Cn + base;

    // stage q_t (per-wave; same-wave LDS ops are in-order, no barrier needed)
    s_q[wave][lane]      = Qb[(size_t)lane * TLEN + t];
    s_q[wave][lane + 32] = Qb[(size_t)(lane + 32) * TLEN + t];

    const int s_lo = (t - RADIUS > 0) ? (t - RADIUS) : 0;
    const int s_hi = (t + RADIUS < TLEN - 1) ? (t + RADIUS) : (TLEN - 1);
    const int L    = s_hi - s_lo + 1;   // <= 101, always >= 1 (s=t in band)

    // --- logits: lane covers band offsets lane, lane+32, lane+64, lane+96
    float lg[4];
    float mx = -3.0e38f;
#pragma unroll
    for (int i = 0; i < 4; ++i) {
        const int sp = i * 32 + lane;
        float v = -3.0e38f;
        if (sp < L) {
            const int s = s_lo + sp;
            const int j = RADIUS + t - s;   // 0..100 inside band
            float a1 = 0.f, a2 = 0.f;
            for (int c = 0; c < DIM; ++c) {
                const float q = s_q[wave][c];
                a1 += q * Kb[(size_t)c * TLEN + s];
                a2 += q * s_emb[j * DIM + c];
            }
            v = a1 + 0.3f * a2;
        }
        lg[i] = v;
        mx = fmaxf(mx, v);
    }

    // --- softmax (wave32 reductions)
#pragma unroll
    for (int off = 16; off; off >>= 1) mx = fmaxf(mx, __shfl_xor(mx, off, 32));

    float p[4];
    float ssum = 0.f;
#pragma unroll
    for (int i = 0; i < 4; ++i) {
        p[i] = (lg[i] > -1.0e37f) ? __expf(lg[i] - mx) : 0.f;
        ssum += p[i];
    }
#pragma unroll
    for (int off = 16; off; off >>= 1) ssum += __shfl_xor(ssum, off, 32);
    const float inv = 1.0f / ssum;

#pragma unroll
    for (int i = 0; i < 4; ++i) s_w[wave][i * 32 + lane] = p[i] * inv;

    // --- output: lanes switch to channel-parallel
#pragma unroll
    for (int ci = 0; ci < 2; ++ci) {
        const int c = lane + ci * 32;
        float a1 = 0.f, a2 = 0.f;
        for (int sp = 0; sp < L; ++sp) {
            const float w = s_w[wave][sp];   // broadcast read
            const int s = s_lo + sp;
            const int j = RADIUS + t - s;
            a1 += w * Cb[(size_t)c * TLEN + s];
            a2 += w * s_emb[j * DIM + c];
        }
        O[base + (size_t)c * TLEN + t] = a1 + 0.3f * a2;
    }
}

// ---------------------------------------------------------------------------
// BatchNorm1d (training stats over (B,T)) + ReLU + scale.  Block per channel.
// ---------------------------------------------------------------------------
__global__ __launch_bounds__(256)
void bn_relu_scale(const float* __restrict__ Yf, const float* __restrict__ gamma,
                   const float* __restrict__ beta, const float* __restrict__ scale,
                   float* __restrict__ out)
{
    __shared__ float s1[256];
    __shared__ float s2[256];
    const int o   = blockIdx.x;
    const int tid = threadIdx.x;

    float sum = 0.f, sq = 0.f;
    for (int i = tid; i < BATCH * TLEN; i += 256) {
        const int bb = i >> 10, t = i & (TLEN - 1);
        const float v = Yf[(size_t)bb * CHANNELS * TLEN + (size_t)o * TLEN + t];
        sum += v;
        sq  += v * v;
    }
    s1[tid] = sum; s2[tid] = sq;
    __syncthreads();
    for (int off = 128; off; off >>= 1) {
        if (tid < off) { s1[tid] += s1[tid + off]; s2[tid] += s2[tid + off]; }
        __syncthreads();
    }
    const float norm = 1.0f / (float)(BATCH * TLEN);
    const float mean = s1[0] * norm;
    const float var  = s2[0] * norm - mean * mean;
    const float g    = gamma[o] * rsqrtf(var + BN_EPS);
    const float bt   = beta[o];
    const float sc   = scale[o];

    for (int i = tid; i < BATCH * TLEN; i += 256) {
        const int bb = i >> 10, t = i & (TLEN - 1);
        const size_t off2 = (size_t)bb * CHANNELS * TLEN + (size_t)o * TLEN + t;
        const float v = (Yf[off2] - mean) * g + bt;
        out[off2] = fmaxf(v, 0.f) * sc;
    }
}

// ---------------------------------------------------------------------------
extern "C" void kernel_launch(void* const* d_in, const int* in_sizes, int n_in,
                              void* d_out, int out_size, void* d_ws, size_t ws_size,
                              hipStream_t stream)
{
    const float* x     = (const float*)d_in[0];
    const float* Wc    = (const float*)d_in[1];
    const float* bc    = (const float*)d_in[2];
    const float* Wq    = (const float*)d_in[3];
    const float* bq    = (const float*)d_in[4];
    const float* Wk    = (const float*)d_in[5];
    const float* bk    = (const float*)d_in[6];
    const float* emb   = (const float*)d_in[7];
    const float* Wfc   = (const float*)d_in[8];
    const float* bfc   = (const float*)d_in[9];
    const float* gamma = (const float*)d_in[10];
    const float* beta  = (const float*)d_in[11];
    const float* scale = (const float*)d_in[12];

    const size_t M = (size_t)BATCH * CHANNELS * TLEN;  // 2,097,152 floats
    float* ws      = (float*)d_ws;                     // needs 5*M*4 = 40 MB
    float* content = ws;
    float* query   = ws + M;
    float* key     = ws + 2 * M;
    float* attout  = ws + 3 * M;
    float* yfc     = ws + 4 * M;

    const dim3 gblk(256);
    const dim3 ggrid(TLEN / 64, CHANNELS / 128, BATCH);

    proj_gemm_wmma<<<ggrid, gblk, 0, stream>>>(Wc, bc, x, content);
    proj_gemm_wmma<<<ggrid, gblk, 0, stream>>>(Wq, bq, x, query);
    proj_gemm_wmma<<<ggrid, gblk, 0, stream>>>(Wk, bk, x, key);

    band_attention<<<dim3(BATCH * HEADS * TLEN / 4), dim3(128), 0, stream>>>(
        query, key, content, emb, attout);

    proj_gemm_wmma<<<ggrid, gblk, 0, stream>>>(Wfc, bfc, attout, yfc);

    bn_relu_scale<<<dim3(CHANNELS), dim3(256), 0, stream>>>(
        yfc, gamma, beta, scale, (float*)d_out);
}